// ResidualVQ_39822936768736
// MI455X (gfx1250) — compile-verified
//
#include <hip/hip_runtime.h>
#include <math.h>

// ---------------- problem constants ----------------
#define BB   16
#define DD   256      // == CD
#define TT   4096
#define NQ   8
#define CS   1024

#define OUT_Q_OFF   ((size_t)0)
#define OUT_Q_N     ((size_t)BB * DD * TT)                  // 16,777,216
#define OUT_IDX_OFF (OUT_Q_N)
#define OUT_IDX_N   ((size_t)NQ * BB * TT)                  // 524,288
#define OUT_COM_OFF (OUT_IDX_OFF + OUT_IDX_N)
#define OUT_COD_OFF (OUT_COM_OFF + NQ)
#define OUT_QS_OFF  (OUT_COD_OFF + NQ)

#define TILE_FLOATS 4096          // one nt tile: 64 k0 * 32 lanes * 2 = 16 KB

typedef __attribute__((ext_vector_type(2))) float v2f;
typedef __attribute__((ext_vector_type(8))) float v8f;

// ---------------------------------------------------------------------------
// Zero the commit/codebook-loss accumulators (atomically accumulated later).
// ---------------------------------------------------------------------------
__global__ void rvq_zero(float* __restrict__ p) {
    if (threadIdx.x < 2 * NQ) p[threadIdx.x] = 0.0f;
}

// ---------------------------------------------------------------------------
// Normalize codebook rows and pack into WMMA B-fragment order.
// B (4x16 KxN, 32-bit): lane l, elem j -> cb_norm[nt*16 + (l%16)][4*k0 + 2*(l/16) + j]
// Packed: cbp[((q*64 + nt)*64 + k0)*32 + lane][j]   (8 MB total in d_ws)
// ---------------------------------------------------------------------------
__global__ __launch_bounds__(128) void rvq_prep(const float* __restrict__ cb,
                                                float* __restrict__ cbp) {
    const int lane = threadIdx.x & 31;
    const int row  = (blockIdx.x << 2) + (threadIdx.x >> 5);   // 0..8191
    const int q    = row >> 10;
    const int n    = row & (CS - 1);

    const float* src = cb + (size_t)row * DD;
    float v[8];
    float ss = 0.0f;
#pragma unroll
    for (int i = 0; i < 8; ++i) {
        v[i] = src[lane + (i << 5)];
        ss += v[i] * v[i];
    }
#pragma unroll
    for (int off = 16; off; off >>= 1) ss += __shfl_xor(ss, off, 32);
    const float sc = 1.0f / fmaxf(sqrtf(ss), 1e-12f);

    const int nt = n >> 4;
#pragma unroll
    for (int i = 0; i < 8; ++i) {
        const int k  = lane + (i << 5);
        const int lp = (n & 15) + (((k >> 1) & 1) << 4);
        const int j  = k & 1;
        const int k0 = k >> 2;
        cbp[((((size_t)q * 64 + nt) * 64 + k0) * 32 + lp) * 2 + j] = v[i] * sc;
    }
}

// ---------------------------------------------------------------------------
// Async-copy one 16 KB fragment tile (global -> LDS), 8 x b128 per thread,
// tracked on ASYNCcnt.  128 threads each move 128 B.
// ---------------------------------------------------------------------------
__device__ __forceinline__ void tile_copy_async(const float* __restrict__ g,
                                                unsigned lds_base, int tid) {
#pragma unroll
    for (int j = 0; j < 8; ++j) {
        const unsigned la = lds_base + (unsigned)(j * 2048 + tid * 16);
        const void*    ga = (const void*)(g + j * 512 + tid * 4);
        asm volatile("global_load_async_to_lds_b128 %0, %1, off"
                     :: "v"(la), "v"(ga) : "memory");
    }
}

// ---------------------------------------------------------------------------
// Fused 8-stage residual VQ.  One wave owns 32 tokens (two A-fragment sets in
// VGPRs for the whole kernel); 4 waves share async-staged LDS code tiles.
// ---------------------------------------------------------------------------
__global__ __launch_bounds__(128) void rvq_main(const float* __restrict__ z,
                                                const float* __restrict__ cb_raw,
                                                const float* __restrict__ cbp,
                                                float* __restrict__ out) {
    float* out_q   = out + OUT_Q_OFF;
    float* out_idx = out + OUT_IDX_OFF;
    float* out_com = out + OUT_COM_OFF;
    float* out_cod = out + OUT_COD_OFF;
    float* out_qs  = out + OUT_QS_OFF;

    __shared__ __align__(16) float buf[2][TILE_FLOATS];   // 2 x 16 KB code tiles
    __shared__ int lds_idx[4][32];

    const int tid  = threadIdx.x;
    const int lane = tid & 31;
    const int wv   = tid >> 5;
    const int bid  = blockIdx.x;                 // 512 blocks
    const int b    = bid >> 5;                   // 32 x 128-token chunks per batch row
    const int t0w  = ((bid & 31) << 7) + (wv << 5);   // wave's first token (32 tokens)
    const int ttk  = lane & 15;                  // A layout: M = lane%16
    const int hh   = lane >> 4;                  // A layout: K-half
    const int kh   = hh << 1;

    const size_t zb = (size_t)b * DD * TT;
    const int    ta = t0w + ttk;                 // set-0 token
    const int    tbk = ta + 16;                  // set-1 token

    const unsigned lb0 = (unsigned)(uintptr_t)(void*)&buf[0][0];
    const unsigned lb1 = (unsigned)(uintptr_t)(void*)&buf[1][0];

    // kick off the async pipeline: tile (q=0, nt=0) -> buf0
    tile_copy_async(cbp, lb0, tid);

    // ---- residual = z, held in VGPRs as two 16x4-f32 A-fragment sets ----
    v2f a0[64], a1[64];
#pragma unroll
    for (int k0 = 0; k0 < 64; ++k0) {
        const int k = 4 * k0 + kh;
        a0[k0].x = z[zb + (size_t)k * TT + ta];
        a0[k0].y = z[zb + (size_t)(k + 1) * TT + ta];
        a1[k0].x = z[zb + (size_t)k * TT + tbk];
        a1[k0].y = z[zb + (size_t)(k + 1) * TT + tbk];
    }

    for (int q = 0; q < NQ; ++q) {
        float bv0[8], bv1[8];
        int   bi0[8], bi1[8];
#pragma unroll
        for (int v = 0; v < 8; ++v) {
            bv0[v] = -__builtin_huge_valf(); bi0[v] = CS;
            bv1[v] = -__builtin_huge_valf(); bi1[v] = CS;
        }

        // ---- score GEMM + running argmax over 64 LDS-staged code tiles ----
        for (int nt = 0; nt < 64; ++nt) {
            const int  tc   = q * 64 + nt;       // global tile counter
            const bool more = !(q == NQ - 1 && nt == 63);
            if (more) {   // prefetch next tile (crosses stage boundary too)
                tile_copy_async(cbp + (size_t)(tc + 1) * TILE_FLOATS,
                                (tc + 1) & 1 ? lb1 : lb0, tid);
                asm volatile("s_wait_asynccnt 8" ::: "memory");
            } else {
                asm volatile("s_wait_asynccnt 0" ::: "memory");
            }
            __syncthreads();                     // tile (q,nt) fully in LDS

            const v2f* lbp = (const v2f*)&buf[tc & 1][0] + lane;
            v8f acc0 = {0.f, 0.f, 0.f, 0.f, 0.f, 0.f, 0.f, 0.f};
            v8f acc1 = {0.f, 0.f, 0.f, 0.f, 0.f, 0.f, 0.f, 0.f};
#pragma unroll
            for (int c = 0; c < 8; ++c) {        // 8 chunks: 8 ds_load_b64 -> 16 WMMAs
                v2f bb[8];
#pragma unroll
                for (int i = 0; i < 8; ++i) bb[i] = lbp[(c * 8 + i) * 32];
#pragma unroll
                for (int i = 0; i < 8; ++i) {
                    acc0 = __builtin_amdgcn_wmma_f32_16x16x4_f32(
                        false, a0[c * 8 + i], false, bb[i], (short)0, acc0, false, false);
                    acc1 = __builtin_amdgcn_wmma_f32_16x16x4_f32(
                        false, a1[c * 8 + i], false, bb[i], (short)0, acc1, false, false);
                }
            }
            // C/D layout: acc[v] = S[M = v + 8*(lane/16)][N = lane%16 + 16*nt]
#pragma unroll
            for (int v = 0; v < 8; ++v) {
                if (acc0[v] > bv0[v]) { bv0[v] = acc0[v]; bi0[v] = (nt << 4) + ttk; }
                if (acc1[v] > bv1[v]) { bv1[v] = acc1[v]; bi1[v] = (nt << 4) + ttk; }
            }
            __syncthreads();                     // all waves done reading this buffer
        }

        // ---- cross-lane argmax within 16-lane halves (first-max wins) ----
#pragma unroll
        for (int v = 0; v < 8; ++v) {
#pragma unroll
            for (int off = 8; off; off >>= 1) {
                float ov = __shfl_xor(bv0[v], off, 16);
                int   oi = __shfl_xor(bi0[v], off, 16);
                if (ov > bv0[v] || (ov == bv0[v] && oi < bi0[v])) { bv0[v] = ov; bi0[v] = oi; }
                ov = __shfl_xor(bv1[v], off, 16);
                oi = __shfl_xor(bi1[v], off, 16);
                if (ov > bv1[v] || (ov == bv1[v] && oi < bi1[v])) { bv1[v] = ov; bi1[v] = oi; }
            }
            if (ttk == 0) {
                const int M = v + (hh << 3);
                lds_idx[wv][M]      = bi0[v];
                lds_idx[wv][16 + M] = bi1[v];
                const size_t io = ((size_t)q * BB + b) * TT + t0w + M;
                out_idx[io]      = (float)bi0[v];
                out_idx[io + 16] = (float)bi1[v];
            }
        }
        asm volatile("s_wait_dscnt 0" ::: "memory");       // LDS RAW within wave
        const int id0 = lds_idx[wv][ttk];
        const int id1 = lds_idx[wv][16 + ttk];

        // ---- gather raw codebook rows, update residual, emit quants[q] ----
        const v2f* cr0 = (const v2f*)(cb_raw + ((size_t)q * CS + id0) * DD);
        const v2f* cr1 = (const v2f*)(cb_raw + ((size_t)q * CS + id1) * DD);
        float* qs = out_qs + (size_t)q * OUT_Q_N + zb;
        float ss = 0.0f;
#pragma unroll
        for (int k0 = 0; k0 < 64; ++k0) {
            const int k  = 4 * k0 + kh;
            const v2f c0 = cr0[2 * k0 + hh];
            const v2f c1 = cr1[2 * k0 + hh];
            a0[k0].x -= c0.x; a0[k0].y -= c0.y;
            a1[k0].x -= c1.x; a1[k0].y -= c1.y;
            ss += a0[k0].x * a0[k0].x + a0[k0].y * a0[k0].y;
            ss += a1[k0].x * a1[k0].x + a1[k0].y * a1[k0].y;
            qs[(size_t)k * TT + ta]        = c0.x;          // coalesced along T
            qs[(size_t)(k + 1) * TT + ta]  = c0.y;
            qs[(size_t)k * TT + tbk]       = c1.x;
            qs[(size_t)(k + 1) * TT + tbk] = c1.y;
        }

        // commit == codebook loss == mean(new_residual^2) over B*D*T
#pragma unroll
        for (int off = 16; off; off >>= 1) ss += __shfl_xor(ss, off, 32);
        if (lane == 0) {
            const float cval = ss * (1.0f / 16777216.0f);   // 1/(B*D*T)
            atomicAdd(&out_com[q], cval);
            atomicAdd(&out_cod[q], cval);
        }
    }

    // ---- quantized_out = z - final residual (== sum of all z_q) ----
#pragma unroll
    for (int k0 = 0; k0 < 64; ++k0) {
        const int k = 4 * k0 + kh;
        out_q[zb + (size_t)k * TT + ta]        = z[zb + (size_t)k * TT + ta]        - a0[k0].x;
        out_q[zb + (size_t)(k + 1) * TT + ta]  = z[zb + (size_t)(k + 1) * TT + ta]  - a0[k0].y;
        out_q[zb + (size_t)k * TT + tbk]       = z[zb + (size_t)k * TT + tbk]       - a1[k0].x;
        out_q[zb + (size_t)(k + 1) * TT + tbk] = z[zb + (size_t)(k + 1) * TT + tbk] - a1[k0].y;
    }
}

// ---------------------------------------------------------------------------
extern "C" void kernel_launch(void* const* d_in, const int* in_sizes, int n_in,
                              void* d_out, int out_size, void* d_ws, size_t ws_size,
                              hipStream_t stream) {
    const float* z  = (const float*)d_in[0];   // [16][256][4096]
    const float* cb = (const float*)d_in[1];   // [8][1024][256]
    float* out = (float*)d_out;
    float* cbp = (float*)d_ws;                 // 8 MB packed normalized codebooks

    rvq_zero<<<1, 32, 0, stream>>>(out + OUT_COM_OFF);
    rvq_prep<<<(NQ * CS) / 4, 128, 0, stream>>>(cb, cbp);
    rvq_main<<<(BB * TT) / 128, 128, 0, stream>>>(z, cb, cbp, out);
}